// MultiBoxLoss_86260123173625
// MI455X (gfx1250) — compile-verified
//
#include <hip/hip_runtime.h>
#include <math.h>

// ---------------- problem constants (match reference) ----------------
#define B_     64
#define P_     16800
#define NOBJ_  32
#define NC_    2
#define THRESH 0.35f
#define VAR0_  0.1f
#define VAR1_  0.2f
#define NEGPOS_ 7

typedef unsigned int       u32;
typedef unsigned long long u64;
typedef __attribute__((ext_vector_type(2))) float v2f;
typedef __attribute__((ext_vector_type(8))) float v8f;

__device__ __forceinline__ float smooth_l1(float d) {
  float a = fabsf(d);
  return a < 1.f ? 0.5f * d * d : a - 0.5f;
}

// ---------------- kernel 0: init scratch ----------------
__global__ void k_init(int* assigned, u64* bp_key, int* num_pos, int* num_pos1,
                       float* loss_c_row, int* has_valid, float* g_scal) {
  int gid = blockIdx.x * blockDim.x + threadIdx.x;
  if (gid < B_ * P_)    assigned[gid] = -1;
  if (gid < B_ * NOBJ_) bp_key[gid]   = 0ull;
  if (gid < B_) { num_pos[gid] = 0; num_pos1[gid] = 0; loss_c_row[gid] = 0.f; has_valid[gid] = 0; }
  if (gid < 4)  g_scal[gid] = 0.f;
}

// ---------------- kernel 1: overlaps + per-prior / per-truth argmax ----------------
// key = (float_bits(iou) << 32) | (~p)  -> atomicMax == argmax with lowest-index tie-break
// (iou >= 0 always, so float bit pattern is order-preserving).
__global__ void k_overlaps(const float* __restrict__ priors, const float* __restrict__ targets,
                           float* __restrict__ bt_over, int* __restrict__ bt_idx,
                           u64* __restrict__ bp_key) {
  __shared__ float tb[NOBJ_][4];
  __shared__ float tarea[NOBJ_];
  __shared__ u64   lkey[NOBJ_];
  int b = blockIdx.y;
  int p = blockIdx.x * blockDim.x + threadIdx.x;
  if (threadIdx.x < NOBJ_) {
    const float* tp = targets + ((size_t)b * NOBJ_ + threadIdx.x) * 15;
    tb[threadIdx.x][0] = tp[0]; tb[threadIdx.x][1] = tp[1];
    tb[threadIdx.x][2] = tp[2]; tb[threadIdx.x][3] = tp[3];
    tarea[threadIdx.x] = (tp[2] - tp[0]) * (tp[3] - tp[1]);
    lkey[threadIdx.x]  = 0ull;
  }
  __syncthreads();
  if (p < P_) {
    float pcx = priors[p*4+0], pcy = priors[p*4+1], pw = priors[p*4+2], ph = priors[p*4+3];
    float px0 = pcx - 0.5f*pw, py0 = pcy - 0.5f*ph, px1 = pcx + 0.5f*pw, py1 = pcy + 0.5f*ph;
    float parea = pw * ph;
    float best = -1.f; int bestt = 0;
    for (int t = 0; t < NOBJ_; ++t) {
      float iw = fmaxf(fminf(tb[t][2], px1) - fmaxf(tb[t][0], px0), 0.f);
      float ih = fmaxf(fminf(tb[t][3], py1) - fmaxf(tb[t][1], py0), 0.f);
      float inter = iw * ih;
      float iou = inter / (tarea[t] + parea - inter);
      if (iou > best) { best = iou; bestt = t; }   // strict > == first-max (jnp.argmax)
      u64 key = ((u64)__float_as_uint(iou) << 32) | (u64)(0xFFFFFFFFu - (u32)p);
      atomicMax(&lkey[t], key);
    }
    bt_over[(size_t)b * P_ + p] = best;
    bt_idx [(size_t)b * P_ + p] = bestt;
  }
  __syncthreads();
  if (threadIdx.x < NOBJ_) atomicMax(&bp_key[b * NOBJ_ + threadIdx.x], lkey[threadIdx.x]);
}

// ---------------- kernel 2: forced matches ----------------
__global__ void k_apply(const u64* __restrict__ bp_key, float* __restrict__ bt_over,
                        int* __restrict__ assigned, int* __restrict__ has_valid) {
  int gid = blockIdx.x * blockDim.x + threadIdx.x;
  if (gid >= B_ * NOBJ_) return;
  int b = gid / NOBJ_, t = gid % NOBJ_;
  u64 key = bp_key[gid];
  float ov = __uint_as_float((u32)(key >> 32));
  int  pidx = (int)(0xFFFFFFFFu - (u32)(key & 0xFFFFFFFFull));
  atomicMax(&assigned[(size_t)b * P_ + pidx], t);       // .at[best_prior_idx].max(t), all t
  if (ov >= 0.2f) {                                     // valid_gt -> force overlap 2.0
    bt_over[(size_t)b * P_ + pidx] = 2.0f;
    atomicOr(&has_valid[b], 1);
  }
}

// ---------------- kernel 3: encode + per-prior losses ----------------
__global__ void k_main(const float* __restrict__ loc_data, const float* __restrict__ conf_data,
                       const float* __restrict__ landm_data, const float* __restrict__ priors,
                       const float* __restrict__ targets,
                       const float* __restrict__ bt_over, const int* __restrict__ bt_idx,
                       const int* __restrict__ assigned, const int* __restrict__ has_valid,
                       float* __restrict__ ce_arr, int* __restrict__ ct_arr,
                       int* num_pos, int* num_pos1, float* g_scal) {
  __shared__ float red[256];
  int b = blockIdx.y;
  int p = blockIdx.x * blockDim.x + threadIdx.x;
  float ll = 0.f, llm = 0.f; int cpos = 0, cpos1 = 0;
  if (p < P_) {
    size_t bp = (size_t)b * P_ + p;
    int a = assigned[bp];
    int tidx = (a >= 0) ? a : bt_idx[bp];
    float over = bt_over[bp];
    const float* tp = targets + ((size_t)b * NOBJ_ + tidx) * 15;
    int conf = (int)tp[14];
    if (over < THRESH)  conf = 0;
    if (!has_valid[b])  conf = 0;
    int pos  = (conf != 0);
    int pos1 = (conf > 0);
    int ct   = pos ? 1 : conf;       // conf_t after where(pos,1,conf_t)
    const float* cd = conf_data + bp * NC_;
    float c0 = cd[0], c1 = cd[1];
    float m   = fmaxf(c0, c1);
    float lse = m + logf(expf(c0 - m) + expf(c1 - m));
    float ce  = lse - (ct == 1 ? c1 : c0);
    ce_arr[bp] = ce; ct_arr[bp] = ct;
    cpos = pos; cpos1 = pos1;
    if (pos | pos1) {
      float pcx = priors[p*4+0], pcy = priors[p*4+1], pw = priors[p*4+2], ph = priors[p*4+3];
      if (pos) {
        float t0 = ((tp[0] + tp[2]) * 0.5f - pcx) / (VAR0_ * pw);
        float t1 = ((tp[1] + tp[3]) * 0.5f - pcy) / (VAR0_ * ph);
        float t2 = logf((tp[2] - tp[0]) / pw) / VAR1_;
        float t3 = logf((tp[3] - tp[1]) / ph) / VAR1_;
        const float* ld = loc_data + bp * 4;
        ll = smooth_l1(ld[0]-t0) + smooth_l1(ld[1]-t1) + smooth_l1(ld[2]-t2) + smooth_l1(ld[3]-t3);
      }
      if (pos1) {
        const float* lm = landm_data + bp * 10;
        #pragma unroll
        for (int j = 0; j < 5; ++j) {
          float gx = (tp[4+2*j]   - pcx) / (VAR0_ * pw);
          float gy = (tp[4+2*j+1] - pcy) / (VAR0_ * ph);
          llm += smooth_l1(lm[2*j] - gx) + smooth_l1(lm[2*j+1] - gy);
        }
      }
    }
  }
  auto breduce = [&](float v) -> float {
    red[threadIdx.x] = v; __syncthreads();
    for (int s = 128; s > 0; s >>= 1) {
      if ((int)threadIdx.x < s) red[threadIdx.x] += red[threadIdx.x + s];
      __syncthreads();
    }
    float r = red[0]; __syncthreads(); return r;
  };
  float sll  = breduce(ll);
  float sllm = breduce(llm);
  float sp   = breduce((float)cpos);
  float sp1  = breduce((float)cpos1);
  if (threadIdx.x == 0) {
    atomicAdd(&g_scal[0], sll);
    atomicAdd(&g_scal[1], sllm);
    atomicAdd(&num_pos [b], (int)(sp  + 0.5f));
    atomicAdd(&num_pos1[b], (int)(sp1 + 0.5f));
  }
}

// ---------------- kernel 4: hard-negative mining (exact radix-select top-k) ----------------
__global__ void k_negmine(const float* __restrict__ ce_arr, const int* __restrict__ ct_arr,
                          const int* __restrict__ num_pos, float* __restrict__ loss_c_row) {
  __shared__ u32 hist[256];
  __shared__ u32 scan[256];
  __shared__ int sh_rem;
  __shared__ u32 sh_byte;
  __shared__ float red[256];
  int b = blockIdx.x, tid = threadIdx.x;
  const int CH = (P_ + 255) / 256;                 // contiguous chunk -> stable index ranks
  int p0 = tid * CH, p1 = (p0 + CH < P_) ? p0 + CH : P_;
  if (p0 > P_) p0 = P_;
  size_t base = (size_t)b * P_;
  long long kk = (long long)NEGPOS_ * num_pos[b];
  if (kk > P_ - 1) kk = P_ - 1;
  int do_sel = kk > 0;
  u32 prefix = 0, hmask = 0;
  int rem = (int)kk;
  if (do_sel) {
    for (int pass = 3; pass >= 0; --pass) {
      int shift = pass * 8;
      hist[tid] = 0; __syncthreads();
      for (int p = p0; p < p1; ++p) {
        int   ct = ct_arr[base + p];
        float lm = (ct != 0) ? 0.f : ce_arr[base + p];   // loss_mine (>= 0)
        u32 key = __float_as_uint(lm);
        if ((key & hmask) == prefix) atomicAdd(&hist[(key >> shift) & 0xFFu], 1u);
      }
      __syncthreads();
      if (tid == 0) {
        int cum = 0, chosen = 0;
        for (int bb = 255; bb >= 0; --bb) {
          int c = (int)hist[bb];
          if (cum + c >= rem) { chosen = bb; break; }
          cum += c;
        }
        sh_rem = rem - cum;          // still needed inside chosen bin
        sh_byte = (u32)chosen;
      }
      __syncthreads();
      rem = sh_rem;
      prefix |= sh_byte << shift;
      hmask  |= 0xFFu   << shift;
      __syncthreads();
    }
  }
  u32 T = prefix;                    // kk-th largest key; rem = #ties to take (by index)
  int eq_local = 0;
  if (do_sel)
    for (int p = p0; p < p1; ++p) {
      int   ct = ct_arr[base + p];
      float lm = (ct != 0) ? 0.f : ce_arr[base + p];
      if (__float_as_uint(lm) == T) eq_local++;
    }
  scan[tid] = (u32)eq_local; __syncthreads();
  if (tid == 0) { u32 acc = 0; for (int i = 0; i < 256; ++i) { u32 c = scan[i]; scan[i] = acc; acc += c; } }
  __syncthreads();
  int eq_base = (int)scan[tid];
  float s = 0.f; int c_eq = 0;
  for (int p = p0; p < p1; ++p) {
    int   ct  = ct_arr[base + p];
    float ce  = ce_arr[base + p];
    int   pos = (ct != 0);
    u32 key = __float_as_uint(pos ? 0.f : ce);
    int sel = pos;                                  // sel = pos | neg
    if (do_sel) {
      if (key > T) sel = 1;
      else if (key == T) { if (eq_base + c_eq < rem) sel = 1; c_eq++; }
    }
    if (sel) s += ce;
  }
  red[tid] = s; __syncthreads();
  for (int st = 128; st > 0; st >>= 1) {
    if (tid < st) red[tid] += red[tid + st];
    __syncthreads();
  }
  if (tid == 0) loss_c_row[b] = red[0];
}

// ---------------- kernel 5: finalize — WMMA 64-way reductions ----------------
// 64 values -> A(16x4 f32), B = ones(4x16): D[m][n] = rowsum(m) for every n, so
// sum64 = sum_{r=0..7} d[r]  (this lane's 8 rows) + same from the partner lane (^16).
__device__ __forceinline__ float wmma_sum64(const float* v, int lane) {
  int m = lane & 15, kb = (lane >> 4) * 2;
  v2f a; a.x = v[m * 4 + kb]; a.y = v[m * 4 + kb + 1];
  v2f ones; ones.x = 1.f; ones.y = 1.f;
  v8f cz = {};
  v8f d = __builtin_amdgcn_wmma_f32_16x16x4_f32(false, a, false, ones,
                                                (short)0, cz, false, false);
  float s = d[0] + d[1] + d[2] + d[3] + d[4] + d[5] + d[6] + d[7];
  return s + __shfl(s, lane ^ 16, 32);   // rows 0-7 + rows 8-15 (wave32)
}

__global__ void k_finalize(const float* __restrict__ loss_c_row,
                           const int* __restrict__ num_pos, const int* __restrict__ num_pos1,
                           const float* __restrict__ g_scal, float* __restrict__ out) {
  __shared__ float fnp[B_], fnp1[B_];
  int lane = threadIdx.x;                          // exactly one wave32, EXEC all ones
  fnp [lane]      = (float)num_pos [lane];
  fnp [lane + 32] = (float)num_pos [lane + 32];
  fnp1[lane]      = (float)num_pos1[lane];
  fnp1[lane + 32] = (float)num_pos1[lane + 32];
  __syncthreads();
  float lossc = wmma_sum64(loss_c_row, lane);
  float Np    = wmma_sum64(fnp,  lane);
  float Np1   = wmma_sum64(fnp1, lane);
  float N  = fmaxf(Np,  1.f);
  float N1 = fmaxf(Np1, 1.f);
  if (lane == 0) {
    out[0] = g_scal[0] / N;     // loss_l / N
    out[1] = lossc     / N;     // loss_c / N
    out[2] = g_scal[1] / N1;    // loss_landm / N1
  }
}

// ---------------- host launcher ----------------
extern "C" void kernel_launch(void* const* d_in, const int* in_sizes, int n_in,
                              void* d_out, int out_size, void* d_ws, size_t ws_size,
                              hipStream_t stream) {
  const float* loc     = (const float*)d_in[0];
  const float* conf    = (const float*)d_in[1];
  const float* landm   = (const float*)d_in[2];
  const float* priors  = (const float*)d_in[3];
  const float* targets = (const float*)d_in[4];
  float* out = (float*)d_out;

  char* ws = (char*)d_ws;
  size_t off = 0;
  auto alloc = [&](size_t bytes) -> char* {
    char* pp = ws + off; off += (bytes + 255) & ~(size_t)255; return pp;
  };
  float* bt_over    = (float*)alloc((size_t)B_ * P_ * 4);
  int*   bt_idx     = (int*)  alloc((size_t)B_ * P_ * 4);
  int*   assigned   = (int*)  alloc((size_t)B_ * P_ * 4);
  float* ce_arr     = (float*)alloc((size_t)B_ * P_ * 4);
  int*   ct_arr     = (int*)  alloc((size_t)B_ * P_ * 4);
  u64*   bp_key     = (u64*)  alloc((size_t)B_ * NOBJ_ * 8);
  int*   num_pos    = (int*)  alloc(B_ * 4);
  int*   num_pos1   = (int*)  alloc(B_ * 4);
  float* loss_c_row = (float*)alloc(B_ * 4);
  int*   has_valid  = (int*)  alloc(B_ * 4);
  float* g_scal     = (float*)alloc(4 * 4);

  int pb = (P_ + 255) / 256;
  k_init<<<dim3((B_ * P_ + 255) / 256), 256, 0, stream>>>(assigned, bp_key, num_pos,
                                                          num_pos1, loss_c_row, has_valid, g_scal);
  k_overlaps<<<dim3(pb, B_), 256, 0, stream>>>(priors, targets, bt_over, bt_idx, bp_key);
  k_apply<<<dim3((B_ * NOBJ_ + 255) / 256), 256, 0, stream>>>(bp_key, bt_over, assigned, has_valid);
  k_main<<<dim3(pb, B_), 256, 0, stream>>>(loc, conf, landm, priors, targets, bt_over, bt_idx,
                                           assigned, has_valid, ce_arr, ct_arr,
                                           num_pos, num_pos1, g_scal);
  k_negmine<<<dim3(B_), 256, 0, stream>>>(ce_arr, ct_arr, num_pos, loss_c_row);
  k_finalize<<<dim3(1), 32, 0, stream>>>(loss_c_row, num_pos, num_pos1, g_scal, out);
}